// Lattice3D_38293928411137
// MI455X (gfx1250) — compile-verified
//
#include <hip/hip_runtime.h>
#include <math.h>

// ---- CDNA5 (gfx1250) lattice MLP using V_WMMA_F32_16X16X4_F32 ----

typedef __attribute__((ext_vector_type(2))) float v2f;
typedef __attribute__((ext_vector_type(8))) float v8f;

#define LDIM   128
#define LMASK  127
#define SSZ    8          // state size
#define HSZ    16         // hidden size
#define NCELL  (LDIM*LDIM*LDIM)
#define NTILES (NCELL/16) // 131072 tiles of 16 cells

#if __has_builtin(__builtin_amdgcn_tanhf)
#define TANHF(x) __builtin_amdgcn_tanhf(x)
#elif __has_builtin(__builtin_amdgcn_tanh_f32)
#define TANHF(x) __builtin_amdgcn_tanh_f32(x)
#else
#define TANHF(x) tanhf(x)
#endif

__global__ __launch_bounds__(256, 1)
void lattice_mlp_wmma_kernel(const float* __restrict__ states,
                             const float* __restrict__ W1,
                             const float* __restrict__ b1,
                             const float* __restrict__ W2,
                             const float* __restrict__ b2,
                             float* __restrict__ out,
                             int tiles_per_wave)
{
    // per-wave private 16x16 transpose patch, padded to stride 20 (bank-conflict free)
    __shared__ float ldsH[8][16 * 20];

    const int lane = threadIdx.x & 31;
    const int wid  = threadIdx.x >> 5;
    const int col  = lane & 15;   // B/C/D column; also A-row (cell within tile)
    const int hi   = lane >> 4;   // half-wave selector

    // ---- preload W1 B-fragments (56x16), 14 K=4 steps ----
    v2f bw1[14];
#pragma unroll
    for (int kk = 0; kk < 14; ++kk) {
        const int k0 = 4 * kk + 2 * hi;
        v2f b;
        b.x = W1[k0 * HSZ + col];
        b.y = W1[(k0 + 1) * HSZ + col];
        bw1[kk] = b;
    }
    // ---- preload W2 B-fragments (16x8 zero-padded to 16x16), 4 K=4 steps ----
    v2f bw2[4];
#pragma unroll
    for (int kk = 0; kk < 4; ++kk) {
        const int k0 = 4 * kk + 2 * hi;
        v2f b;
        b.x = (col < SSZ) ? W2[k0 * SSZ + col] : 0.0f;
        b.y = (col < SSZ) ? W2[(k0 + 1) * SSZ + col] : 0.0f;
        bw2[kk] = b;
    }
    const float bias1 = b1[col];
    const float bias2 = (col < SSZ) ? b2[col] : 0.0f;

    const int gwave = blockIdx.x * 8 + wid;
    float* hrow = &ldsH[wid][0];

    for (int t = 0; t < tiles_per_wave; ++t) {
        const int tile = gwave * tiles_per_wave + t;
        const int cell = (tile << 4) + col;         // this lane's A-matrix row (cell)
        const int x = cell & LMASK;
        const int y = (cell >> 7) & LMASK;
        const int z = (cell >> 14) & LMASK;
        const int rowb = cell & ~LMASK;             // (z,y) fixed, x varies
        const int zx   = (z << 14) + x;             // (z,x) fixed, y varies
        const int yx   = (y << 7) + x;              // (y,x) fixed, z varies

        // 7 neighbor-block byte offsets, feature order: own,x-1,x+1,y-1,y+1,z-1,z+1
        unsigned off0 = (unsigned)cell << 5;
        unsigned off1 = (unsigned)(rowb + ((x + LMASK) & LMASK)) << 5;
        unsigned off2 = (unsigned)(rowb + ((x + 1) & LMASK)) << 5;
        unsigned off3 = (unsigned)(zx + (((y + LMASK) & LMASK) << 7)) << 5;
        unsigned off4 = (unsigned)(zx + (((y + 1) & LMASK) << 7)) << 5;
        unsigned off5 = (unsigned)((((z + LMASK) & LMASK) << 14) + yx) << 5;
        unsigned off6 = (unsigned)((((z + 1) & LMASK) << 14) + yx) << 5;
        unsigned nboff[7] = {off0, off1, off2, off3, off4, off5, off6};

        // ---- layer 1: acc(16x16) = inp(16x56) * W1(56x16) + b1 ----
        v8f acc;
#pragma unroll
        for (int i = 0; i < 8; ++i) acc[i] = bias1;

#pragma unroll
        for (int kk = 0; kk < 14; ++kk) {
            // A frag: lane holds inp[cell][4kk+2hi .. +1]; stays inside one 8-float block
            const unsigned byteoff =
                nboff[kk >> 1] + (unsigned)(((kk & 1) << 4) + (hi << 3));
            v2f a = *(const v2f*)((const char*)states + byteoff);
            acc = __builtin_amdgcn_wmma_f32_16x16x4_f32(
                false, a, false, bw1[kk], (short)0, acc, false, false);
        }

        // ---- tanh (hardware v_tanh_f32 when available) ----
#pragma unroll
        for (int i = 0; i < 8; ++i) acc[i] = TANHF(acc[i]);

        // ---- transpose D-layout -> A-layout through per-wave LDS patch ----
#pragma unroll
        for (int i = 0; i < 8; ++i)
            hrow[(i + (hi << 3)) * 20 + col] = acc[i];

        asm volatile("s_wait_dscnt 0" ::: "memory");  // cross-lane RAW inside wave

        // ---- layer 2: acc2(16x16) = h(16x16) * W2pad(16x16) + b2 ----
        v8f acc2;
#pragma unroll
        for (int i = 0; i < 8; ++i) acc2[i] = bias2;

#pragma unroll
        for (int kk = 0; kk < 4; ++kk) {
            v2f a2 = *(const v2f*)&hrow[col * 20 + 4 * kk + 2 * hi];
            acc2 = __builtin_amdgcn_wmma_f32_16x16x4_f32(
                false, a2, false, bw2[kk], (short)0, acc2, false, false);
        }

        // ---- store: rows = cells (vgpr index + 8*hi), cols 0..7 = state ----
        if (col < SSZ) {
            const unsigned obase = ((unsigned)tile << 9) + ((unsigned)hi << 8)
                                 + ((unsigned)col << 2);
#pragma unroll
            for (int i = 0; i < 8; ++i)
                *(float*)((char*)out + (obase + ((unsigned)i << 5))) = acc2[i];
        }
    }
}

extern "C" void kernel_launch(void* const* d_in, const int* in_sizes, int n_in,
                              void* d_out, int out_size, void* d_ws, size_t ws_size,
                              hipStream_t stream) {
    const float* states = (const float*)d_in[0];
    const float* W1     = (const float*)d_in[1];
    const float* b1     = (const float*)d_in[2];
    const float* W2     = (const float*)d_in[3];
    const float* b2     = (const float*)d_in[4];
    float* out = (float*)d_out;

    const int blocks = 2048;                       // 2048 * 8 waves = 16384 waves
    const int tpw    = NTILES / (blocks * 8);      // = 8 tiles per wave, exact
    lattice_mlp_wmma_kernel<<<blocks, 256, 0, stream>>>(
        states, W1, b1, W2, b2, out, tpw);
}